// SS2D_76690936038060
// MI455X (gfx1250) — compile-verified
//
#include <hip/hip_runtime.h>
#include <math.h>

typedef __attribute__((ext_vector_type(2))) float v2f;
typedef __attribute__((ext_vector_type(8))) float v8f;

// Problem constants
#define L_TOT   4096      // 64*64
#define DM      96        // D_MODEL
#define DI      192       // D_INNER
#define KDIR    4
#define DSTATE  16
#define DTRANK  6
#define NCROWS  38        // DT_RANK + 2*D_STATE
#define KD_L    786432    // DI*L_TOT
#define NCHUNK  64        // scan chunks along L
#define CHLEN   64        // L_TOT / NCHUNK
#define LN_EPS  1e-5f

// gfx1250 async global->LDS path (guarded; falls back to plain LDS stores)
#if defined(__has_builtin)
#if __has_builtin(__builtin_amdgcn_global_load_async_to_lds_b32)
#define HAVE_ASYNC_LDS 1
#endif
#endif

__device__ __forceinline__ void wait_async_then_barrier() {
#ifdef HAVE_ASYNC_LDS
#if __has_builtin(__builtin_amdgcn_s_wait_asynccnt)
  __builtin_amdgcn_s_wait_asynccnt(0);
#else
  asm volatile("s_wait_asynccnt 0" ::: "memory");
#endif
#endif
  __syncthreads();
}

__device__ __forceinline__ void lds_fill_one(const float* __restrict__ gp,
                                             float* lp) {
#ifdef HAVE_ASYNC_LDS
  __builtin_amdgcn_global_load_async_to_lds_b32(
      (__attribute__((address_space(1))) int*)gp,
      (__attribute__((address_space(3))) int*)lp, 0, 0);
#else
  *lp = *gp;
#endif
}

// ---------------------------------------------------------------------------
// Kernel 1: xz = x @ in_proj_w^T     (M=L=4096, N=384, K=96)  -- f32 WMMA
// xq (e<192) stored d-major [d][l] for conv; z (e>=192) stored [l][192].
// ---------------------------------------------------------------------------
__global__ __launch_bounds__(32) void k_inproj_wmma(
    const float* __restrict__ x, const float* __restrict__ w,
    float* __restrict__ xq, float* __restrict__ z) {
  const int lane = threadIdx.x;
  const int half = lane >> 4;        // 0: lanes 0-15, 1: lanes 16-31
  const int lm   = lane & 15;
  const int l0   = blockIdx.x * 16;  // M tile (spatial)
  const int e0   = blockIdx.y * 16;  // N tile (channel)
  v8f acc = {};
  for (int k0 = 0; k0 < DM; k0 += 4) {
    v2f a, b;
    const float* xr = x + (l0 + lm) * DM + k0 + 2 * half;
    a.x = xr[0]; a.y = xr[1];
    const float* wr = w + (e0 + lm) * DM + k0 + 2 * half;
    b.x = wr[0]; b.y = wr[1];
    acc = __builtin_amdgcn_wmma_f32_16x16x4_f32(false, a, false, b,
                                                (short)0, acc, false, false);
  }
#pragma unroll
  for (int i = 0; i < 8; ++i) {
    const int l = l0 + i + half * 8;
    const int e = e0 + lm;
    const float v = acc[i];
    if (e < DI) xq[e * L_TOT + l] = v;
    else        z[l * DI + (e - DI)] = v;
  }
}

// ---------------------------------------------------------------------------
// Kernel 2: depthwise 3x3 conv + bias + SiLU, fan out into 4 scan directions.
// ---------------------------------------------------------------------------
__global__ void k_conv_silu(const float* __restrict__ xq,
                            const float* __restrict__ cw,
                            const float* __restrict__ cb,
                            float* __restrict__ xs, float* __restrict__ us) {
  const int idx = blockIdx.x * blockDim.x + threadIdx.x;
  if (idx >= DI * L_TOT) return;
  const int d = idx >> 12;
  const int l = idx & (L_TOT - 1);
  const int h = l >> 6, w = l & 63;
  const float* wk = cw + d * 9;
  float acc = cb[d];
#pragma unroll
  for (int i = 0; i < 3; ++i)
#pragma unroll
    for (int j = 0; j < 3; ++j) {
      const int hh = h + i - 1, ww = w + j - 1;
      if (hh >= 0 && hh < 64 && ww >= 0 && ww < 64)
        acc += xq[d * L_TOT + hh * 64 + ww] * wk[i * 3 + j];
    }
  const float v = acc / (1.0f + __expf(-acc));   // SiLU
  const int lt = w * 64 + h;                      // transposed scan order
  xs[0 * KD_L + d * L_TOT + l]            = v;
  xs[1 * KD_L + d * L_TOT + lt]           = v;
  xs[2 * KD_L + d * L_TOT + (4095 - l)]   = v;
  xs[3 * KD_L + d * L_TOT + (4095 - lt)]  = v;
  us[(0 * L_TOT + l)           * DI + d] = v;
  us[(1 * L_TOT + lt)          * DI + d] = v;
  us[(2 * L_TOT + (4095 - l))  * DI + d] = v;
  us[(3 * L_TOT + (4095 - lt)) * DI + d] = v;
}

// ---------------------------------------------------------------------------
// Kernel 3: x_dbl[k][c][l] = sum_d x_proj_w[k][c][d] * xs[k][d][l]  -- WMMA
// ---------------------------------------------------------------------------
__global__ __launch_bounds__(32) void k_xdbl_wmma(
    const float* __restrict__ xs, const float* __restrict__ xpw,
    float* __restrict__ xdbl) {
  const int lane = threadIdx.x;
  const int half = lane >> 4;
  const int lm   = lane & 15;
  const int l0   = blockIdx.x * 16;
  const int c0   = blockIdx.y * 16;
  const int k    = blockIdx.z;
  const float* xsk = xs  + (size_t)k * KD_L;
  const float* wk  = xpw + (size_t)k * NCROWS * DI;
  v8f acc = {};
  for (int d0 = 0; d0 < DI; d0 += 4) {
    v2f a, b;
    const int c  = c0 + lm;
    const int ka = d0 + 2 * half;
    const bool cv = (c < NCROWS);
    a.x = cv ? wk[c * DI + ka]     : 0.0f;
    a.y = cv ? wk[c * DI + ka + 1] : 0.0f;
    const int l = l0 + lm;
    b.x = xsk[(size_t)(d0 + 2 * half)     * L_TOT + l];
    b.y = xsk[(size_t)(d0 + 2 * half + 1) * L_TOT + l];
    acc = __builtin_amdgcn_wmma_f32_16x16x4_f32(false, a, false, b,
                                                (short)0, acc, false, false);
  }
#pragma unroll
  for (int i = 0; i < 8; ++i) {
    const int c = c0 + i + half * 8;
    const int l = l0 + lm;
    if (c < NCROWS)
      xdbl[((size_t)k * NCROWS + c) * L_TOT + l] = acc[i];
  }
}

// ---------------------------------------------------------------------------
// Kernel 4: delta[k][l][d] = softplus( dt_w[k][d][:] . x_dbl[k][0:6][l] + dt_b )
// ---------------------------------------------------------------------------
__global__ void k_delta(const float* __restrict__ xdbl,
                        const float* __restrict__ dtw,
                        const float* __restrict__ dtb,
                        float* __restrict__ delta) {
  const int idx = blockIdx.x * blockDim.x + threadIdx.x;
  if (idx >= KDIR * L_TOT * DI) return;
  const int k   = idx / (L_TOT * DI);
  const int rem = idx - k * (L_TOT * DI);
  const int l   = rem / DI;
  const int d   = rem - l * DI;
  const float* xd = xdbl + (size_t)k * NCROWS * L_TOT;
  const float* wt = dtw + ((size_t)k * DI + d) * DTRANK;
  float s = dtb[k * DI + d];
#pragma unroll
  for (int r = 0; r < DTRANK; ++r)
    s += wt[r] * xd[(size_t)r * L_TOT + l];
  delta[idx] = (s > 20.0f) ? s : log1pf(__expf(s));  // softplus
}

// ---------------------------------------------------------------------------
// Blocked selective scan (3 phases).  h[l] = a[l]*h[l-1] + b[l] is linear, so
// each 64-step chunk reduces to a transition (aProd, hLocal); a tiny pass
// scans the 64 summaries; chunks then replay with the correct incoming state.
// Phases 1/3 run 4*64 blocks x 192 threads (1536 waves) instead of 24 waves.
// Summary layout: [k][chunk][d][n].
// ---------------------------------------------------------------------------
__global__ __launch_bounds__(192) void k_scan_phase1(
    const float* __restrict__ us, const float* __restrict__ delta,
    const float* __restrict__ xdbl, const float* __restrict__ A_logs,
    float* __restrict__ aP, float* __restrict__ hP) {
  const int ch = blockIdx.x;
  const int k  = blockIdx.y;
  const int d  = threadIdx.x;
  const int l0 = ch * CHLEN;
  __shared__ float sB[CHLEN][DSTATE];     // B rows 6..21 for this chunk (4 KB)

  const float* xdk = xdbl + (size_t)k * NCROWS * L_TOT;
  for (int e = d; e < CHLEN * DSTATE; e += DI) {
    const int n  = e & 15;
    const int ll = e >> 4;
    lds_fill_one(xdk + (size_t)(6 + n) * L_TOT + (l0 + ll), &sB[ll][n]);
  }
  float A[DSTATE], h[DSTATE], ap[DSTATE];
#pragma unroll
  for (int n = 0; n < DSTATE; ++n) {
    A[n]  = -__expf(A_logs[((size_t)k * DI + d) * DSTATE + n]);
    h[n]  = 0.0f;
    ap[n] = 1.0f;
  }
  wait_async_then_barrier();

  const size_t base = (size_t)k * L_TOT * DI;
#pragma unroll 4
  for (int ll = 0; ll < CHLEN; ++ll) {
    const int l = l0 + ll;
    const float dlt = delta[base + (size_t)l * DI + d];
    const float du  = dlt * us[base + (size_t)l * DI + d];
#pragma unroll
    for (int n = 0; n < DSTATE; ++n) {
      const float a = __expf(dlt * A[n]);
      ap[n] *= a;
      h[n] = a * h[n] + du * sB[ll][n];
    }
  }
  const size_t so = (((size_t)k * NCHUNK + ch) * DI + d) * DSTATE;
#pragma unroll
  for (int n = 0; n < DSTATE; ++n) {
    aP[so + n] = ap[n];
    hP[so + n] = h[n];
  }
}

__global__ void k_scan_phase2(const float* __restrict__ aP,
                              const float* __restrict__ hP,
                              float* __restrict__ hI) {
  const int idx = blockIdx.x * blockDim.x + threadIdx.x;   // (k,d,n)
  if (idx >= KDIR * DI * DSTATE) return;
  const int k  = idx / (DI * DSTATE);
  const int dn = idx - k * (DI * DSTATE);
  float h = 0.0f;
  for (int c = 0; c < NCHUNK; ++c) {
    const size_t o = ((size_t)(k * NCHUNK + c) * DI * DSTATE) + dn;
    hI[o] = h;                          // exclusive: state entering chunk c
    h = aP[o] * h + hP[o];
  }
}

__global__ __launch_bounds__(192) void k_scan_phase3(
    const float* __restrict__ us, const float* __restrict__ delta,
    const float* __restrict__ xdbl, const float* __restrict__ A_logs,
    const float* __restrict__ Ds, const float* __restrict__ hI,
    float* __restrict__ ys) {
  const int ch = blockIdx.x;
  const int k  = blockIdx.y;
  const int d  = threadIdx.x;
  const int l0 = ch * CHLEN;
  __shared__ float sBC[CHLEN][2 * DSTATE];   // B and C rows 6..37 (8 KB)

  const float* xdk = xdbl + (size_t)k * NCROWS * L_TOT;
  for (int e = d; e < CHLEN * 2 * DSTATE; e += DI) {
    const int n  = e & 31;
    const int ll = e >> 5;
    lds_fill_one(xdk + (size_t)(6 + n) * L_TOT + (l0 + ll), &sBC[ll][n]);
  }
  float A[DSTATE], h[DSTATE];
  const size_t so = (((size_t)k * NCHUNK + ch) * DI + d) * DSTATE;
#pragma unroll
  for (int n = 0; n < DSTATE; ++n) {
    A[n] = -__expf(A_logs[((size_t)k * DI + d) * DSTATE + n]);
    h[n] = hI[so + n];
  }
  const float Dd = Ds[k * DI + d];
  wait_async_then_barrier();

  const size_t base = (size_t)k * L_TOT * DI;
#pragma unroll 4
  for (int ll = 0; ll < CHLEN; ++ll) {
    const int l = l0 + ll;
    const float dlt = delta[base + (size_t)l * DI + d];
    const float uu  = us[base + (size_t)l * DI + d];
    const float du  = dlt * uu;
    float y = 0.0f;
#pragma unroll
    for (int n = 0; n < DSTATE; ++n) {
      const float a = __expf(dlt * A[n]);
      h[n] = a * h[n] + du * sBC[ll][n];
      y += h[n] * sBC[ll][DSTATE + n];
    }
    ys[base + (size_t)l * DI + d] = y + Dd * uu;
  }
}

// ---------------------------------------------------------------------------
// Kernel 6: combine 4 directions + LayerNorm over d + SiLU(z) gate.
// ---------------------------------------------------------------------------
__global__ __launch_bounds__(192) void k_combine_ln(
    const float* __restrict__ ys, const float* __restrict__ z,
    const float* __restrict__ ln_g, const float* __restrict__ ln_b,
    float* __restrict__ yb) {
  const int l = blockIdx.x;
  const int d = threadIdx.x;
  const int h = l >> 6, w = l & 63;
  const int lt = w * 64 + h;
  float v = ys[((size_t)(0 * L_TOT) + l)           * DI + d]
          + ys[((size_t)(2 * L_TOT) + (4095 - l))  * DI + d]
          + ys[((size_t)(1 * L_TOT) + lt)          * DI + d]
          + ys[((size_t)(3 * L_TOT) + (4095 - lt)) * DI + d];

  __shared__ float red[DI];
  red[d] = v;
  __syncthreads();
  for (int s = 96; s >= 3; s >>= 1) {
    if (d < s) red[d] += red[d + s];
    __syncthreads();
  }
  const float mu = (red[0] + red[1] + red[2]) * (1.0f / DI);
  __syncthreads();
  const float c = v - mu;
  red[d] = c * c;
  __syncthreads();
  for (int s = 96; s >= 3; s >>= 1) {
    if (d < s) red[d] += red[d + s];
    __syncthreads();
  }
  const float var = (red[0] + red[1] + red[2]) * (1.0f / DI);
  const float yn  = c * rsqrtf(var + LN_EPS) * ln_g[d] + ln_b[d];
  const float zv  = z[(size_t)l * DI + d];
  yb[(size_t)l * DI + d] = yn * (zv / (1.0f + __expf(-zv)));
}

// ---------------------------------------------------------------------------
// Kernel 7: out = yb @ out_proj_w^T   (M=L=4096, N=96, K=192) -- f32 WMMA
// ---------------------------------------------------------------------------
__global__ __launch_bounds__(32) void k_outproj_wmma(
    const float* __restrict__ yb, const float* __restrict__ opw,
    float* __restrict__ out) {
  const int lane = threadIdx.x;
  const int half = lane >> 4;
  const int lm   = lane & 15;
  const int l0   = blockIdx.x * 16;
  const int c0   = blockIdx.y * 16;
  v8f acc = {};
  for (int d0 = 0; d0 < DI; d0 += 4) {
    v2f a, b;
    const float* ar = yb + (size_t)(l0 + lm) * DI + d0 + 2 * half;
    a.x = ar[0]; a.y = ar[1];
    const float* br = opw + (size_t)(c0 + lm) * DI + d0 + 2 * half;
    b.x = br[0]; b.y = br[1];
    acc = __builtin_amdgcn_wmma_f32_16x16x4_f32(false, a, false, b,
                                                (short)0, acc, false, false);
  }
#pragma unroll
  for (int i = 0; i < 8; ++i) {
    const int l = l0 + i + half * 8;
    const int c = c0 + lm;
    out[(size_t)l * DM + c] = acc[i];
  }
}

// ---------------------------------------------------------------------------
extern "C" void kernel_launch(void* const* d_in, const int* in_sizes, int n_in,
                              void* d_out, int out_size, void* d_ws, size_t ws_size,
                              hipStream_t stream) {
  const float* x    = (const float*)d_in[0];
  const float* ipw  = (const float*)d_in[1];
  const float* cw   = (const float*)d_in[2];
  const float* cb   = (const float*)d_in[3];
  const float* xpw  = (const float*)d_in[4];
  const float* dtw  = (const float*)d_in[5];
  const float* dtb  = (const float*)d_in[6];
  const float* alog = (const float*)d_in[7];
  const float* ds   = (const float*)d_in[8];
  const float* lng  = (const float*)d_in[9];
  const float* lnb  = (const float*)d_in[10];
  const float* opw  = (const float*)d_in[11];
  float* out = (float*)d_out;

  // workspace carving (floats)
  float* ws    = (float*)d_ws;
  float* xq    = ws;                                 // DI*L
  float* z     = xq    + KD_L;                       // L*DI
  float* xs    = z     + KD_L;                       // 4*DI*L
  float* us    = xs    + 4 * KD_L;                   // 4*L*DI
  float* xdbl  = us    + 4 * KD_L;                   // 4*38*L
  float* delta = xdbl  + KDIR * NCROWS * L_TOT;      // 4*L*DI
  float* ysb   = delta + 4 * KD_L;                   // 4*L*DI
  float* yb    = ysb   + 4 * KD_L;                   // L*DI
  float* aP    = yb    + KD_L;                       // 4*64*DI*16
  float* hP    = aP    + KDIR * NCHUNK * DI * DSTATE;
  float* hI    = hP    + KDIR * NCHUNK * DI * DSTATE;

  // 1) in_proj GEMM (WMMA f32)
  k_inproj_wmma<<<dim3(L_TOT / 16, (2 * DI) / 16), 32, 0, stream>>>(x, ipw, xq, z);
  // 2) depthwise conv + SiLU + direction fan-out
  k_conv_silu<<<(DI * L_TOT + 255) / 256, 256, 0, stream>>>(xq, cw, cb, xs, us);
  // 3) x_proj GEMM per direction (WMMA f32)
  k_xdbl_wmma<<<dim3(L_TOT / 16, 3, KDIR), 32, 0, stream>>>(xs, xpw, xdbl);
  // 4) dt low-rank + softplus
  k_delta<<<(KDIR * L_TOT * DI + 255) / 256, 256, 0, stream>>>(xdbl, dtw, dtb, delta);
  // 5) blocked selective scan (async LDS staging)
  k_scan_phase1<<<dim3(NCHUNK, KDIR), DI, 0, stream>>>(us, delta, xdbl, alog, aP, hP);
  k_scan_phase2<<<(KDIR * DI * DSTATE + 255) / 256, 256, 0, stream>>>(aP, hP, hI);
  k_scan_phase3<<<dim3(NCHUNK, KDIR), DI, 0, stream>>>(us, delta, xdbl, alog, ds, hI, ysb);
  // 6) combine directions + LayerNorm + gate
  k_combine_ln<<<L_TOT, DI, 0, stream>>>(ysb, z, lng, lnb, yb);
  // 7) out_proj GEMM (WMMA f32)
  k_outproj_wmma<<<dim3(L_TOT / 16, DM / 16), 32, 0, stream>>>(yb, opw, out);
}